// Sub_Hist_TCN_38096359916149
// MI455X (gfx1250) — compile-verified
//
#include <hip/hip_runtime.h>
#include <hip/hip_bf16.h>

// ---------------- problem constants ----------------
#define SUB_NO 20
#define E_NO   2000
#define T_NO   50
#define HID_NO 10
#define SH     (SUB_NO * HID_NO)   // 200
#define BATCH  8
#define TLEN   5000

typedef float v2f __attribute__((ext_vector_type(2)));
typedef float v8f __attribute__((ext_vector_type(8)));

// ---------------------------------------------------------------------------
// Kernel 1: A'[s,e] = C_syn_e[s,e] * exp(E_scale[e]), zero-padded to 32 rows
// ---------------------------------------------------------------------------
__global__ void scaleA_kernel(const float* __restrict__ C_syn_e,
                              const float* __restrict__ E_scale,
                              float* __restrict__ A32) {
    int i = blockIdx.x * blockDim.x + threadIdx.x;   // 0 .. 32*E_NO-1
    if (i >= 32 * E_NO) return;
    int s = i / E_NO;
    int e = i - s * E_NO;
    A32[i] = (s < SUB_NO) ? C_syn_e[s * E_NO + e] * __expf(E_scale[e]) : 0.0f;
}

// ---------------------------------------------------------------------------
// Kernel 2: syn[b,s,t] = sum_e A'[s,e] * S_e[b,t,e]  via V_WMMA_F32_16X16X4_F32
// One wave per (b, 16-wide t tile). Two M tiles (rows 0-15 / 16-31, rows>=20
// are zero padding in A') share one B fragment per K step.
//
// Tail handling: a lane whose time index tn >= TLEN has its row pointer
// clamped to row 0 (valid, finite memory). Its B fragment feeds ONLY output
// column N = lane&15 (B rows are striped across lanes), and exactly those
// columns fail the `t < TLEN` store guard — so no masking/branching is
// needed inside the K loop, and EXEC stays all-1s for every WMMA.
// ---------------------------------------------------------------------------
__global__ __launch_bounds__(32) void gemm_syn_kernel(
        const float* __restrict__ S_e,   // [B, T, E]
        const float* __restrict__ A32,   // [32, E] (padded, pre-scaled)
        float* __restrict__ syn) {       // [B, SUB, T]
    const int lane = threadIdx.x;        // wave32
    const int t0   = blockIdx.x * 16;
    const int b    = blockIdx.y;

    const int m    = lane & 15;          // A row / B col within tile
    const int kb   = (lane >> 4) * 2;    // K pair: lanes 0-15 -> {0,1}, 16-31 -> {2,3}

    const float* __restrict__ A0 = A32 + (size_t)m * E_NO;          // M rows 0..15
    const float* __restrict__ A1 = A32 + (size_t)(m + 16) * E_NO;   // M rows 16..31

    const int tn      = t0 + m;                                     // B column (time)
    const int tn_safe = (tn < TLEN) ? tn : 0;                       // clamp (see note)
    const float* __restrict__ Brow =
        S_e + ((size_t)b * TLEN + (size_t)tn_safe) * E_NO;

    v8f acc0 = {};
    v8f acc1 = {};

    for (int k0 = 0; k0 < E_NO; k0 += 4) {
        const int k = k0 + kb;
        v2f a0; a0.x = A0[k];   a0.y = A0[k + 1];
        v2f a1; a1.x = A1[k];   a1.y = A1[k + 1];
        v2f bf; bf.x = Brow[k]; bf.y = Brow[k + 1];
        __builtin_prefetch(Brow + k + 512, 0, 1);   // global_prefetch_b8
        // 8 args: (neg_a, A, neg_b, B, c_mod, C, reuse_a, reuse_b)
        acc0 = __builtin_amdgcn_wmma_f32_16x16x4_f32(
                   false, a0, false, bf, (short)0, acc0, false, false);
        acc1 = __builtin_amdgcn_wmma_f32_16x16x4_f32(
                   false, a1, false, bf, (short)0, acc1, false, false);
    }

    // C/D layout: element r of lane L -> row M = r + 8*(L>=16), col N = L&15
    const int half = lane >> 4;
    const int n    = lane & 15;
    const int t    = t0 + n;
    if (t < TLEN) {
        #pragma unroll
        for (int r = 0; r < 8; ++r) {
            int s0 = r + 8 * half;            // rows of first tile
            if (s0 < SUB_NO)
                syn[((size_t)b * SUB_NO + s0) * TLEN + t] = acc0[r];
            int s1 = 16 + r + 8 * half;       // rows of second tile
            if (s1 < SUB_NO)
                syn[((size_t)b * SUB_NO + s1) * TLEN + t] = acc1[r];
        }
    }
}

// ---------------------------------------------------------------------------
// Kernel 3: causal grouped conv:
//   conv[b,c,t] = sum_{j=0..49} W_e_layer1[c,j] * syn[b, c/HID, t-j]
// ---------------------------------------------------------------------------
__global__ __launch_bounds__(256) void conv_kernel(
        const float* __restrict__ syn,   // [B, SUB, T]
        const float* __restrict__ W1,    // [SH, T_NO]
        float* __restrict__ conv) {      // [B, SH, T]
    const int t = blockIdx.x * blockDim.x + threadIdx.x;
    const int c = blockIdx.y;
    const int b = blockIdx.z;
    if (t >= TLEN) return;
    const int s = c / HID_NO;
    const float* __restrict__ sr = syn + ((size_t)b * SUB_NO + s) * TLEN;
    const float* __restrict__ w  = W1 + (size_t)c * T_NO;
    float acc = 0.0f;
    const int jmax = (t < T_NO - 1) ? t : (T_NO - 1);
    for (int j = 0; j <= jmax; ++j)
        acc = fmaf(w[j], sr[t - j], acc);
    conv[((size_t)b * SH + c) * TLEN + t] = acc;
}

// ---------------------------------------------------------------------------
// Kernel 4: sequential AR(50)+tanh scan. One thread per channel (b fixed per
// block, c = tid < 200). 50-deep circular history window lives in LDS
// (dynamic LDS indexing; per-thread column -> no bank conflicts, no barriers
// needed inside the time loop). Emits ys * exp(W_layer2[c]) directly.
// ---------------------------------------------------------------------------
__global__ __launch_bounds__(256) void scan_kernel(
        const float* __restrict__ conv,   // [B, SH, T]
        const float* __restrict__ W_hist, // [SH, T_NO]
        const float* __restrict__ b1,     // [SH]
        const float* __restrict__ W2,     // [SUB, HID] flat == [SH]
        float* __restrict__ yw) {         // [B, SH, T]
    __shared__ float win[T_NO][256];      // 50 KB LDS
    const int tid = threadIdx.x;
    const int b   = blockIdx.x;

    #pragma unroll
    for (int j = 0; j < T_NO; ++j) win[j][tid] = 0.0f;
    __syncthreads();

    if (tid < SH) {
        const int c = tid;
        float wreg[T_NO];
        #pragma unroll
        for (int k = 0; k < T_NO; ++k) wreg[k] = W_hist[(size_t)c * T_NO + k];
        const float bias = b1[c];
        const float w2   = __expf(W2[c]);
        const float* __restrict__ cv  = conv + ((size_t)b * SH + c) * TLEN;
        float*       __restrict__ out = yw   + ((size_t)b * SH + c) * TLEN;

        int pos = 0;   // write slot for ys[t]  (== t % 50)
        for (int t = 0; t < TLEN; ++t) {
            float acc = cv[t] + bias;
            #pragma unroll
            for (int d = 1; d <= T_NO; ++d) {
                int idx = pos - d;                 // slot of ys[t-d]
                idx += (idx < 0) ? T_NO : 0;
                acc = fmaf(wreg[d - 1], win[idx][tid], acc);
            }
            const float y = tanhf(acc);
            win[pos][tid] = y;
            out[t] = y * w2;
            pos = (pos + 1 == T_NO) ? 0 : pos + 1;
        }
    }
}

// ---------------------------------------------------------------------------
// Kernel 5: out[b,t] = V_o + sum_c yw[b,c,t]
// ---------------------------------------------------------------------------
__global__ __launch_bounds__(256) void reduce_kernel(
        const float* __restrict__ yw,
        const float* __restrict__ V_o,
        float* __restrict__ out) {
    const int i = blockIdx.x * blockDim.x + threadIdx.x;  // b*TLEN + t
    if (i >= BATCH * TLEN) return;
    const int b = i / TLEN;
    const int t = i - b * TLEN;
    const float* __restrict__ base = yw + (size_t)b * SH * TLEN + t;
    float acc = 0.0f;
    for (int c = 0; c < SH; ++c)
        acc += base[(size_t)c * TLEN];
    out[i] = acc + V_o[0];
}

// ---------------------------------------------------------------------------
extern "C" void kernel_launch(void* const* d_in, const int* in_sizes, int n_in,
                              void* d_out, int out_size, void* d_ws, size_t ws_size,
                              hipStream_t stream) {
    const float* S_e     = (const float*)d_in[0];
    // d_in[1] = S_i (unused)
    const float* C_syn_e = (const float*)d_in[2];
    // d_in[3] = C_syn_i (unused)
    const float* E_scale = (const float*)d_in[4];
    // d_in[5] = I_scale (unused)
    const float* W1      = (const float*)d_in[6];   // W_e_layer1 [SH, T_NO]
    const float* W2      = (const float*)d_in[7];   // W_layer2 [SUB, HID]
    const float* b1      = (const float*)d_in[8];   // b_layer1 [SH]
    const float* W_hist  = (const float*)d_in[9];   // [SH, T_NO]
    // d_in[10] = W_sub (unused)
    const float* V_o     = (const float*)d_in[11];  // [1]
    float* out = (float*)d_out;

    // workspace layout (floats)
    float* ws    = (float*)d_ws;
    float* A32   = ws;                                   //   64,000 f
    float* syn   = ws + 64000;                           //  800,000 f
    float* conv  = ws + 64000 + 800000;                  // 8,000,000 f
    float* yw    = ws + 64000 + 800000 + 8000000;        // 8,000,000 f

    // 1) pre-scale + pad A
    scaleA_kernel<<<(32 * E_NO + 255) / 256, 256, 0, stream>>>(C_syn_e, E_scale, A32);

    // 2) WMMA GEMM: syn[b,s,t]
    dim3 ggrid((TLEN + 15) / 16, BATCH);
    gemm_syn_kernel<<<ggrid, 32, 0, stream>>>(S_e, A32, syn);

    // 3) causal grouped conv
    dim3 cgrid((TLEN + 255) / 256, SH, BATCH);
    conv_kernel<<<cgrid, 256, 0, stream>>>(syn, W1, conv);

    // 4) sequential scan (one block per batch element)
    scan_kernel<<<BATCH, 256, 0, stream>>>(conv, W_hist, b1, W2, yw);

    // 5) reduction over channels + V_o
    reduce_kernel<<<(BATCH * TLEN + 255) / 256, 256, 0, stream>>>(yw, V_o, out);
}